// Gate_66211215835147
// MI455X (gfx1250) — compile-verified
//
#include <hip/hip_runtime.h>
#include <hip/hip_bf16.h>
#include <math.h>

typedef __attribute__((ext_vector_type(2))) float v2f;
typedef __attribute__((ext_vector_type(8))) float v8f;

#define B_    64
#define T_    577
#define D_    768
#define DENS_ 519   // int(577*0.9)
#define NSKIP_ 58   // 577 - 519

// ---------------------------------------------------------------------------
// Kernel 1: prob = sigmoid(x . w + b) via V_WMMA_F32_16X16X4_F32.
// One wave (32 lanes) scores 16 token rows. A-matrix layout (ISA 7.12.2):
//   lane L<16  : A[M=L][K=k+0], A[M=L][K=k+1] in the two A VGPRs
//   lane L>=16 : A[M=L-16][K=k+2], A[M=L-16][K=k+3]
// B is the matching w chunk broadcast over all 16 columns, so every column of
// the 16x16 C tile holds the same per-row dot product. Lane 0 / lane 16 own
// M=0..7 / M=8..15 of column N=0 (C/D layout) and write the results.
// ---------------------------------------------------------------------------
__global__ __launch_bounds__(32)
void score_kernel(const float* __restrict__ x, const float* __restrict__ w,
                  const float* __restrict__ bias, float* __restrict__ prob,
                  int nrows) {
  const int lane = threadIdx.x & 31;
  const int half = lane >> 4;        // which K pair this lane-half holds
  const int m    = lane & 15;        // row within the 16-row tile
  long row = (long)blockIdx.x * 16 + m;
  long rc  = row < nrows ? row : (long)nrows - 1;   // clamp for safe loads
  const float* xr = x + rc * D_ + half * 2;
  const float* wr = w + half * 2;

  v8f c = {0.f, 0.f, 0.f, 0.f, 0.f, 0.f, 0.f, 0.f};
#pragma unroll 4
  for (int k = 0; k < D_; k += 4) {
    v2f a;  a.x  = xr[k]; a.y  = xr[k + 1];
    v2f bb; bb.x = wr[k]; bb.y = wr[k + 1];
    c = __builtin_amdgcn_wmma_f32_16x16x4_f32(false, a, false, bb,
                                              (short)0, c, false, false);
  }

  if (m == 0) {                       // lanes 0 and 16: column N=0 of C
    float bv = bias[0];
    long base = (long)blockIdx.x * 16 + half * 8;
#pragma unroll
    for (int j = 0; j < 8; ++j) {     // VGPR j of C = row (half*8 + j)
      long r = base + j;
      if (r < nrows) {
        float z = c[j] + bv;
        prob[r] = 1.0f / (1.0f + __expf(-z));
      }
    }
  }
}

// ---------------------------------------------------------------------------
// Kernel 2: exact stable descending rank-sort of 577 probs per batch.
// Keys are positive floats (sigmoid output), so raw bit compare is monotonic.
// Tie-break (j < t) matches top_k's stable ascending-index rule.
// Skip list = reversed tail of the descending order; softmax weights computed
// with the standard max-subtraction (max of skipped = sorted position DENS_).
// ---------------------------------------------------------------------------
__global__ __launch_bounds__(640)
void sort_kernel(const float* __restrict__ prob, int* __restrict__ sidx_g,
                 float* __restrict__ wts_g) {
  __shared__ float sp[T_];
  __shared__ int   si[T_];
  __shared__ float se[NSKIP_];
  __shared__ float ssum;
  const int b = blockIdx.x;
  const int t = threadIdx.x;

  if (t < T_) sp[t] = prob[(long)b * T_ + t];
  __syncthreads();

  if (t < T_) {
    unsigned mk = __float_as_uint(sp[t]);
    int rank = 0;
    for (int j = 0; j < T_; ++j) {
      unsigned kj = __float_as_uint(sp[j]);
      rank += (kj > mk) || (kj == mk && j < t);
    }
    si[rank] = t;
  }
  __syncthreads();

  if (t < T_) sidx_g[(long)b * T_ + t] = si[t];

  float pmax = sp[si[DENS_]];         // largest prob among the skipped 58
  if (t < NSKIP_) {
    float p = sp[si[T_ - 1 - t]];     // ascending-prob order (top_k of -prob)
    se[t] = __expf(p - pmax);
  }
  __syncthreads();
  if (t == 0) {
    float s = 0.f;
    for (int j = 0; j < NSKIP_; ++j) s += se[j];
    ssum = s;
  }
  __syncthreads();
  if (t < NSKIP_) wts_g[(long)b * NSKIP_ + t] = se[t] / ssum;
}

// ---------------------------------------------------------------------------
// Kernel 3: row gather. One block per (row, batch); 192 threads x float4 move
// a 768-float row. r < DENS_ -> tokens region; else reversed-tail skip region.
// ---------------------------------------------------------------------------
__global__ __launch_bounds__(192)
void gather_kernel(const float* __restrict__ x, const int* __restrict__ sidx,
                   float* __restrict__ out) {
  const int r = blockIdx.x;           // 0..576
  const int b = blockIdx.y;           // 0..63
  const int pos = (r < DENS_) ? r : (T_ - 1 - (r - DENS_));
  const int idx = sidx[(long)b * T_ + pos];
  const float4* src = (const float4*)(x + ((long)b * T_ + idx) * D_);
  float4* dst;
  if (r < DENS_)
    dst = (float4*)(out + ((long)b * DENS_ + r) * D_);
  else
    dst = (float4*)(out + (long)B_ * DENS_ * D_ +
                    ((long)b * NSKIP_ + (r - DENS_)) * D_);
  dst[threadIdx.x] = src[threadIdx.x];
}

// ---------------------------------------------------------------------------
// Kernel 4: summary_token[b] = sum_j weights[b][j] * skip_tokens[b][j].
// Skip rows are L2-hot after the gather. 192 threads x float4 per batch.
// ---------------------------------------------------------------------------
__global__ __launch_bounds__(192)
void summary_kernel(const float* __restrict__ x, const int* __restrict__ sidx,
                    const float* __restrict__ wts, float* __restrict__ out) {
  const int b = blockIdx.x;
  const int t = threadIdx.x;
  float4 acc = {0.f, 0.f, 0.f, 0.f};
  for (int j = 0; j < NSKIP_; ++j) {
    int idx  = sidx[(long)b * T_ + (T_ - 1 - j)];
    float wj = wts[(long)b * NSKIP_ + j];
    const float4* src = (const float4*)(x + ((long)b * T_ + idx) * D_);
    float4 v = src[t];
    acc.x += wj * v.x; acc.y += wj * v.y;
    acc.z += wj * v.z; acc.w += wj * v.w;
  }
  float4* dst = (float4*)(out + (long)B_ * T_ * D_ + (long)b * D_);
  dst[t] = acc;
}

extern "C" void kernel_launch(void* const* d_in, const int* in_sizes, int n_in,
                              void* d_out, int out_size, void* d_ws,
                              size_t ws_size, hipStream_t stream) {
  (void)in_sizes; (void)n_in; (void)out_size; (void)ws_size;
  const float* x    = (const float*)d_in[0];
  const float* w    = (const float*)d_in[1];
  const float* bias = (const float*)d_in[2];
  float* out = (float*)d_out;

  float* prob = (float*)d_ws;                 // B*T floats
  int*   sidx = (int*)(prob + (long)B_ * T_); // B*T ints
  float* wts  = (float*)(sidx + (long)B_ * T_); // B*NSKIP floats

  const int nrows = B_ * T_;
  score_kernel<<<(nrows + 15) / 16, 32, 0, stream>>>(x, w, bias, prob, nrows);
  sort_kernel<<<B_, 640, 0, stream>>>(prob, sidx, wts);
  dim3 ggrid(T_, B_);
  gather_kernel<<<ggrid, 192, 0, stream>>>(x, sidx, out);
  summary_kernel<<<B_, 192, 0, stream>>>(x, sidx, wts, out);
}